// KernelActivation_32006096290235
// MI455X (gfx1250) — compile-verified
//
#include <hip/hip_runtime.h>

// Patch-softmax over 2x2 neighborhoods of x[16,64,256,256] fp32.
//
// Geometry (fixed by the reference):
//   planes      = 16*64 = 1024 independent 256x256 images
//   patch rows  = 128 per plane (pairs of image rows)
//   each thread = 4 consecutive floats of row 2r and of row 2r+1
//                 -> two independent 2x2 softmax patches, 128-bit in/out.
//
// Memory-bound: 512 MB total traffic -> ~22 us at 23.3 TB/s. All accesses are
// 16B per lane, contiguous across the wave (512B/wave/access), non-temporal
// (data is single-touch and larger than the 192 MB L2).

#define W        256
#define PLANE_SZ (256 * 256)   // 65536 floats per (b,c) plane
#define THREADS_PER_BLOCK 256

// Native clang vector type: required by __builtin_nontemporal_load/store
// (HIP's float4 is a struct and is rejected by the builtin).
typedef float v4f __attribute__((ext_vector_type(4)));

__global__ __launch_bounds__(THREADS_PER_BLOCK) void
patch_softmax2x2_kernel(const float* __restrict__ x, float* __restrict__ y) {
    const unsigned tid = blockIdx.x * THREADS_PER_BLOCK + threadIdx.x;

    // chunk : which group of 4 columns (64 per row)
    // prow  : which patch-row (128 per plane)
    // plane : which (b,c) image (1024 total)
    const unsigned chunk = tid & 63u;
    const unsigned prow  = (tid >> 6) & 127u;
    const unsigned plane = tid >> 13;

    const size_t base = (size_t)plane * PLANE_SZ + (size_t)(prow * 2u) * W + chunk * 4u;

    // 128-bit coalesced non-temporal loads: row 2r and row 2r+1.
    const v4f a = __builtin_nontemporal_load((const v4f*)(x + base));
    const v4f b = __builtin_nontemporal_load((const v4f*)(x + base + W));

    // ---- patch 0: {a.x, a.y, b.x, b.y} ----
    float m0 = fmaxf(fmaxf(a.x, a.y), fmaxf(b.x, b.y));
    float e00 = __expf(a.x - m0);
    float e01 = __expf(a.y - m0);
    float e02 = __expf(b.x - m0);
    float e03 = __expf(b.y - m0);
    // sum >= 1 (max element contributes exp(0)=1) -> rcp is well conditioned
    float r0 = __builtin_amdgcn_rcpf((e00 + e01) + (e02 + e03));

    // ---- patch 1: {a.z, a.w, b.z, b.w} ----
    float m1 = fmaxf(fmaxf(a.z, a.w), fmaxf(b.z, b.w));
    float e10 = __expf(a.z - m1);
    float e11 = __expf(a.w - m1);
    float e12 = __expf(b.z - m1);
    float e13 = __expf(b.w - m1);
    float r1 = __builtin_amdgcn_rcpf((e10 + e11) + (e12 + e13));

    v4f o0, o1;
    o0.x = e00 * r0;  o0.y = e01 * r0;  o0.z = e10 * r1;  o0.w = e11 * r1;
    o1.x = e02 * r0;  o1.y = e03 * r0;  o1.z = e12 * r1;  o1.w = e13 * r1;

    // 128-bit coalesced non-temporal stores.
    __builtin_nontemporal_store(o0, (v4f*)(y + base));
    __builtin_nontemporal_store(o1, (v4f*)(y + base + W));
}

extern "C" void kernel_launch(void* const* d_in, const int* in_sizes, int n_in,
                              void* d_out, int out_size, void* d_ws, size_t ws_size,
                              hipStream_t stream) {
    (void)n_in; (void)out_size; (void)d_ws; (void)ws_size;
    const float* x = (const float*)d_in[0];
    float*       y = (float*)d_out;

    // 8 elements (two 128-bit rows of one patch-pair) per thread.
    const int n_elems   = in_sizes[0];                   // 67,108,864
    const int n_threads = n_elems / 8;                   // 8,388,608
    const int n_blocks  = n_threads / THREADS_PER_BLOCK; // 32,768

    patch_softmax2x2_kernel<<<n_blocks, THREADS_PER_BLOCK, 0, stream>>>(x, y);
}